// RotatedIoU3DLoss_57947698758132
// MI455X (gfx1250) — compile-verified
//
#include <hip/hip_runtime.h>
#include <hip/hip_bf16.h>
#include <stdint.h>
#include <math.h>

#define BLOCK 256
#define NCAND 24

// ---------------------------------------------------------------------------
// gfx1250 async global->LDS staging (CDNA5 ASYNCcnt path), guarded so the
// kernel still compiles if the builtin is absent / named differently.
// Probe result (round 1): builtin exists; param 1 is a generic, non-const
// 'int *'. Pass plain 'int*' for both pointers; codegen inserts addrspacecasts.
// ---------------------------------------------------------------------------
#if defined(__AMDGCN__) && defined(__HIP_DEVICE_COMPILE__)
#  if __has_builtin(__builtin_amdgcn_global_load_async_to_lds_b32)
#    define HAVE_ASYNC_LDS 1
#    define ASYNC_CP_B32(gSrc, ldsDst)                                         \
       __builtin_amdgcn_global_load_async_to_lds_b32(                          \
           (int*)(gSrc), (int*)(ldsDst), 0, 0)
#  elif __has_builtin(__builtin_amdgcn_load_to_lds)
#    define HAVE_ASYNC_LDS 1
#    define ASYNC_CP_B32(gSrc, ldsDst)                                         \
       __builtin_amdgcn_load_to_lds((void*)(gSrc),                             \
           (__attribute__((address_space(3))) void*)(ldsDst), 4, 0, 0)
#  endif
#endif

__device__ __forceinline__ void wait_async_zero() {
#if defined(HAVE_ASYNC_LDS)
#  if __has_builtin(__builtin_amdgcn_s_wait_asynccnt)
  __builtin_amdgcn_s_wait_asynccnt(0);
#  else
  asm volatile("s_wait_asynccnt 0" ::: "memory");
#  endif
#endif
}

__device__ __forceinline__ float fast_rcp(float x) {
#if defined(__HIP_DEVICE_COMPILE__)
  return __builtin_amdgcn_rcpf(x);
#else
  return 1.0f / x;
#endif
}

// Monotone surrogate for atan2 on (-pi, pi]: "diamond angle" in (-2, 2].
// Only the sort ORDER matters, so this replaces ~25-op atan2 with ~5 ops.
__device__ __forceinline__ float pseudo_angle(float dx, float dy) {
  float ax = fabsf(dx) + fabsf(dy);
  float p  = 1.0f - dx * fast_rcp(ax);   // [0,2], increasing with |angle|
  p = (ax == 0.0f) ? 0.0f : p;
  return copysignf(p, dy);               // sign from dy (matches atan2 branch)
}

// ---------------------------------------------------------------------------
// Compile-time Batcher odd-even merge sorting network for arbitrary n=24.
// All compare-exchange indices are constants after unrolling -> the 24-entry
// (key,x,y) arrays stay in VGPRs (no scratch), each CE is min/max + cndmask.
// ---------------------------------------------------------------------------
struct SortNet {
  int cnt;
  unsigned char a[192];
  unsigned char b[192];
};

static constexpr SortNet buildNet(int n) {
  SortNet s{};
  s.cnt = 0;
  for (int p = 1; p < n; p *= 2)
    for (int k = p; k >= 1; k /= 2)
      for (int j = k % p; j + k <= n - 1; j += 2 * k) {
        int imax = (k - 1 < n - j - k - 1) ? (k - 1) : (n - j - k - 1);
        for (int i = 0; i <= imax; ++i)
          if ((i + j) / (2 * p) == (i + j + k) / (2 * p)) {
            s.a[s.cnt] = (unsigned char)(i + j);
            s.b[s.cnt] = (unsigned char)(i + j + k);
            ++s.cnt;
          }
      }
  return s;
}
static constexpr SortNet NET = buildNet(NCAND);

__device__ __forceinline__ void box2corners(float x, float y, float w, float h,
                                            float ang, float* cx, float* cy) {
  float s, c;
  __sincosf(ang, &s, &c);
  const float hx[4] = {0.5f, -0.5f, -0.5f, 0.5f};
  const float hy[4] = {0.5f, 0.5f, -0.5f, -0.5f};
#pragma unroll
  for (int k = 0; k < 4; ++k) {
    float dx = hx[k] * w, dy = hy[k] * h;
    cx[k] = dx * c - dy * s + x;
    cy[k] = dx * s + dy * c + y;
  }
}

// corners (cx1,cy1) inside oriented box given by (cx2,cy2): project onto AB/AD.
__device__ __forceinline__ void box_in_box(const float* cx1, const float* cy1,
                                           const float* cx2, const float* cy2,
                                           bool* cond) {
  float ax = cx2[0], ay = cy2[0];
  float abx = cx2[1] - ax, aby = cy2[1] - ay;
  float adx = cx2[3] - ax, ady = cy2[3] - ay;
  float iab = fast_rcp(abx * abx + aby * aby);
  float iad = fast_rcp(adx * adx + ady * ady);
#pragma unroll
  for (int k = 0; k < 4; ++k) {
    float amx = cx1[k] - ax, amy = cy1[k] - ay;
    float rab = (abx * amx + aby * amy) * iab;
    float rad = (adx * amx + ady * amy) * iad;
    cond[k] = (rab > -1e-6f) && (rab < 1.0f + 1e-6f) &&
              (rad > -1e-6f) && (rad < 1.0f + 1e-6f);
  }
}

__global__ __launch_bounds__(BLOCK) void rotated_iou3d_loss_kernel(
    const float* __restrict__ pred, const float* __restrict__ tgt,
    float* __restrict__ out, int n) {
  __shared__ float lp[BLOCK * 7];
  __shared__ float lt[BLOCK * 7];

  const int tid = threadIdx.x;
  const long long base = (long long)blockIdx.x * BLOCK;
  const float* gp = pred + base * 7;
  const float* gt = tgt + base * 7;

  __builtin_prefetch(gp, 0, 0);   // -> global_prefetch_b8
  __builtin_prefetch(gt, 0, 0);

  if (base + BLOCK <= n) {
#if defined(HAVE_ASYNC_LDS)
#pragma unroll
    for (int k = 0; k < 7; ++k) {
      const int idx = k * BLOCK + tid;     // coalesced dword stream
      ASYNC_CP_B32(gp + idx, &lp[idx]);
      ASYNC_CP_B32(gt + idx, &lt[idx]);
    }
    wait_async_zero();
#else
#pragma unroll
    for (int k = 0; k < 7; ++k) {
      const int idx = k * BLOCK + tid;
      lp[idx] = gp[idx];
      lt[idx] = gt[idx];
    }
#endif
  } else {
    const long long lim = (long long)n * 7 - base * 7;
    for (int k = 0; k < 7; ++k) {
      const int idx = k * BLOCK + tid;
      if (idx < lim) { lp[idx] = gp[idx]; lt[idx] = gt[idx]; }
    }
  }
  __syncthreads();

  const long long box = base + tid;
  if (box >= n) return;

  const float* P = &lp[tid * 7];   // stride 7 is coprime with 64 banks
  const float* T = &lt[tid * 7];
  float p0 = P[0], p1 = P[1], p2 = P[2], p3 = P[3], p4 = P[4], p5 = P[5], p6 = P[6];
  float t0 = T[0], t1 = T[1], t2_ = T[2], t3 = T[3], t4 = T[4], t5 = T[5], t6 = T[6];

  // ---- 2D rotated corners ----
  float cx1[4], cy1[4], cx2[4], cy2[4];
  box2corners(p0, p1, p3, p4, p6, cx1, cy1);
  box2corners(t0, t1, t3, t4, t6, cx2, cy2);

  // ---- candidate vertices: 4+4 contained corners + 16 edge intersections ----
  float vx[NCAND], vy[NCAND], key[NCAND];
  bool msk[NCAND];

  bool c12[4], c21[4];
  box_in_box(cx1, cy1, cx2, cy2, c12);
  box_in_box(cx2, cy2, cx1, cy1, c21);
#pragma unroll
  for (int k = 0; k < 4; ++k) {
    vx[k] = cx1[k]; vy[k] = cy1[k]; msk[k] = c12[k];
    vx[4 + k] = cx2[k]; vy[4 + k] = cy2[k]; msk[4 + k] = c21[k];
  }

  float e1x[4], e1y[4], e2x[4], e2y[4];
#pragma unroll
  for (int i = 0; i < 4; ++i) {
    int ip = (i + 1) & 3;
    e1x[i] = cx1[i] - cx1[ip]; e1y[i] = cy1[i] - cy1[ip];
    e2x[i] = cx2[i] - cx2[ip]; e2y[i] = cy2[i] - cy2[ip];
  }
#pragma unroll
  for (int i = 0; i < 4; ++i) {
#pragma unroll
    for (int j = 0; j < 4; ++j) {
      float x1 = cx1[i], y1 = cy1[i];
      float dx13 = x1 - cx2[j], dy13 = y1 - cy2[j];
      float num   = e1x[i] * e2y[j] - e1y[i] * e2x[j];
      float den_t = dx13 * e2y[j] - dy13 * e2x[j];
      float den_u = e1x[i] * dy13 - e1y[i] * dx13;
      bool  nz  = (num != 0.0f);
      float inv = fast_rcp(num);
      float t   = den_t * inv;
      float u   = -(den_u * inv);
      bool  m   = nz && (t > 0.0f) && (t < 1.0f) && (u > 0.0f) && (u < 1.0f);
      float tt  = den_t * fast_rcp(num + 1e-8f);
      int   s_  = 8 + i * 4 + j;
      vx[s_] = m ? (x1 - tt * e1x[i]) : 0.0f;   // masked pts are zero (as ref)
      vy[s_] = m ? (y1 - tt * e1y[i]) : 0.0f;
      msk[s_] = m;
    }
  }

  // ---- centroid over masked vertices, pseudo-angle keys ----
  int nvert = 0;
  float sx = 0.0f, sy = 0.0f;
#pragma unroll
  for (int i = 0; i < NCAND; ++i) {
    nvert += msk[i] ? 1 : 0;
    sx += msk[i] ? vx[i] : 0.0f;
    sy += msk[i] ? vy[i] : 0.0f;
  }
  float invn = fast_rcp((float)(nvert > 0 ? nvert : 1));
  float mx = sx * invn, my = sy * invn;
#pragma unroll
  for (int i = 0; i < NCAND; ++i)
    key[i] = msk[i] ? pseudo_angle(vx[i] - mx, vy[i] - my) : __builtin_inff();

  // ---- register-resident sorting network (static indices -> no scratch) ----
#pragma unroll
  for (int c = 0; c < NET.cnt; ++c) {
    const int ia = NET.a[c], ib = NET.b[c];
    float ka = key[ia], kb = key[ib];
    bool sw = kb < ka;
    key[ia] = fminf(ka, kb);
    key[ib] = fmaxf(ka, kb);
    float xa = vx[ia], xb = vx[ib];
    float ya = vy[ia], yb = vy[ib];
    vx[ia] = sw ? xb : xa; vx[ib] = sw ? xa : xb;
    vy[ia] = sw ? yb : ya; vy[ib] = sw ? ya : yb;
  }

  // ---- masked shoelace, closing the polygon at v[0] ----
  float area2 = 0.0f;
#pragma unroll
  for (int i = 0; i < NCAND; ++i) {
    const int ip = (i + 1) % NCAND;
    bool wrap = (i + 1 >= nvert);
    float qx = wrap ? vx[0] : vx[ip];
    float qy = wrap ? vy[0] : vy[ip];
    float cr = vx[i] * qy - vy[i] * qx;
    area2 += (i < nvert) ? cr : 0.0f;
  }
  float inter2d = 0.5f * fabsf(area2);

  // ---- z-overlap, volumes, IoU ----
  float zmax1 = p2 + 0.5f * p5, zmin1 = p2 - 0.5f * p5;
  float zmax2 = t2_ + 0.5f * t5, zmin2 = t2_ - 0.5f * t5;
  float zov = fmaxf(fminf(zmax1, zmax2) - fmaxf(zmin1, zmin2), 0.0f);
  float inter3d = inter2d * zov;
  float vol1 = p3 * p4 * p5;
  float vol2 = t3 * t4 * t5;
  float uni = vol1 + vol2 - inter3d;
  float iou = inter3d / uni;

  out[box] = 1.0f - iou;   // LOSS_WEIGHT == 1.0
}

extern "C" void kernel_launch(void* const* d_in, const int* in_sizes, int n_in,
                              void* d_out, int out_size, void* d_ws, size_t ws_size,
                              hipStream_t stream) {
  (void)n_in; (void)d_ws; (void)ws_size; (void)out_size;
  const float* pred = (const float*)d_in[0];
  const float* tgt  = (const float*)d_in[1];
  float* out = (float*)d_out;
  const int n = in_sizes[0] / 7;
  const int blocks = (n + BLOCK - 1) / BLOCK;
  hipLaunchKernelGGL(rotated_iou3d_loss_kernel, dim3(blocks), dim3(BLOCK), 0,
                     stream, pred, tgt, out, n);
}